// EnergyWell_60043642798467
// MI455X (gfx1250) — compile-verified
//
#include <hip/hip_runtime.h>
#include <math.h>

#define BLOCK 256
#define C_WELLS 64

typedef int   v4i __attribute__((ext_vector_type(4)));
typedef float v4f __attribute__((ext_vector_type(4)));

__global__ __launch_bounds__(BLOCK) void EnergyWell_kernel(
    const float* __restrict__ inst_pos,     // N*2
    const float* __restrict__ half_sizes,   // N*2
    const float* __restrict__ well_boxes,   // 64*4 : xl,yl,xh,yh
    const int*   __restrict__ avail,        // N*64
    const float* __restrict__ exponents,    // N
    float*       __restrict__ out,          // N
    int N)
{
    // 64 boxes * 16B = 1KB staged in LDS
    __shared__ __align__(16) float s_box[C_WELLS * 4];

    const int tid = threadIdx.x;

    // ---- CDNA5 async global->LDS staging (ASYNCcnt-tracked) ----
    // lanes 0..63 (waves 0 and 1) each move one 16-byte box row.
    if (tid < C_WELLS) {
        unsigned lds_addr = (unsigned)(uintptr_t)(&s_box[tid * 4]);
        const float* gptr = well_boxes + tid * 4;
        asm volatile("global_load_async_to_lds_b128 %0, %1, off"
                     :
                     : "v"(lds_addr), "v"(gptr)
                     : "memory");
        asm volatile("s_wait_asynccnt 0x0" ::: "memory");
    }
    __syncthreads();

    const int i = blockIdx.x * BLOCK + tid;
    if (i >= N) return;

    // per-instance scalars (coalesced b64 loads)
    const float2 p = ((const float2*)inst_pos)[i];
    const float2 h = ((const float2*)half_sizes)[i];
    const float xlo = p.x - h.x, xhi = p.x + h.x;
    const float ylo = p.y - h.y, yhi = p.y + h.y;

    float best = INFINITY;
    int   bidx = 0;

    // 256 contiguous bytes per lane, streamed with non-temporal b128 loads
    const v4i* __restrict__ arow = (const v4i*)(avail + (size_t)i * C_WELLS);

#pragma unroll
    for (int c4 = 0; c4 < C_WELLS / 4; ++c4) {
        const v4i a = __builtin_nontemporal_load(arow + c4);
#pragma unroll
        for (int k = 0; k < 4; ++k) {
            const int c = c4 * 4 + k;
            const v4f b = ((const v4f*)s_box)[c];           // ds_load_b128 broadcast
            const float dx = fmaxf(fmaxf(b.x - xlo, xhi - b.z), 0.0f);
            const float dy = fmaxf(fmaxf(b.y - ylo, yhi - b.w), 0.0f);
            const float d  = dx + dy;
            // avail && strictly-less: 2x v_cmp -> s_and (SALU) -> 2x cndmask.
            // strict '<' preserves first-occurrence argmin tie semantics.
            const bool lt = (a[k] != 0) & (d < best);
            best = lt ? d : best;
            bidx = lt ? c : bidx;
        }
    }

    // recompute dx,dy only for the winning well (divergent LDS read)
    const v4f b = ((const v4f*)s_box)[bidx];
    const float dx = fmaxf(fmaxf(b.x - xlo, xhi - b.z), 0.0f);
    const float dy = fmaxf(fmaxf(b.y - ylo, yhi - b.w), 0.0f);

    const float e = exponents[i];
    float energy;
    if (e == 2.0f) {
        energy = dx * dx + dy * dy;        // fast path: harness uses e = 2.0
    } else {
        energy = powf(dx, e) + powf(dy, e);
    }
    __builtin_nontemporal_store(energy, out + i);
}

extern "C" void kernel_launch(void* const* d_in, const int* in_sizes, int n_in,
                              void* d_out, int out_size, void* d_ws, size_t ws_size,
                              hipStream_t stream) {
    // setup_inputs() order:
    // 0: inst_pos (N*2 f32)        1: half_inst_sizes (N*2 f32)
    // 2: inst_areas (N f32, UNUSED) 3: well_boxes (64*4 f32)
    // 4: inst_cr_avail_map (N*64 int) 5: exponents (N f32)
    const float* inst_pos   = (const float*)d_in[0];
    const float* half_sizes = (const float*)d_in[1];
    const float* well_boxes = (const float*)d_in[3];
    const int*   avail      = (const int*)d_in[4];
    const float* exponents  = (const float*)d_in[5];
    float*       out        = (float*)d_out;

    const int N = in_sizes[5];  // exponents has one entry per instance
    const int grid = (N + BLOCK - 1) / BLOCK;
    EnergyWell_kernel<<<grid, BLOCK, 0, stream>>>(
        inst_pos, half_sizes, well_boxes, avail, exponents, out, N);
}